// OrthogonalHouseholderAlternative_1580547968091
// MI455X (gfx1250) — compile-verified
//
#include <hip/hip_runtime.h>

// OrthogonalHouseholderAlternative for MI455X (gfx1250), fp32 WMMA path.
//
// d_in[0] = x [65536,1024] f32, d_in[1] = A [1024,1024] f32, d_in[2] = b [1024] f32
// d_out   = out [65536,1024] f32
// d_ws    : B (1M f32) | p (1M f32) | inv2diag (1024 f32); pT reuses B's slot.

typedef __attribute__((ext_vector_type(2))) float v2f;
typedef __attribute__((ext_vector_type(8))) float v8f;

#define SZ 1024
#define SXP 1028   // padded LDS row stride (floats): stride%64==4 -> conflict-free lane banks

__device__ __forceinline__ v8f wmma4(v2f a, v2f b, v8f c) {
  // D(16x16,f32) = A(16x4,f32) x B(4x16,f32) + C
  return __builtin_amdgcn_wmma_f32_16x16x4_f32(false, a, false, b, (short)0, c,
                                               false, false);
}

// ---- CDNA5 async global->LDS (ASYNCcnt-tracked, bypasses VGPRs) ----
// ISA 15.18.3 opcode 98: GLOBAL_LOAD_ASYNC_TO_LDS_B128 vdst(LDS byte offset), vaddr, off
__device__ __forceinline__ void async_copy_b128(void* lds_dst, const void* gsrc) {
  const unsigned loff = (unsigned)(unsigned long long)lds_dst;   // LDS byte offset
  const unsigned long long ga = (unsigned long long)gsrc;        // global address
  asm volatile("global_load_async_to_lds_b128 %0, %1, off"
               :: "v"(loff), "v"(ga) : "memory");
}

__device__ __forceinline__ void wait_async0() {
#if __has_builtin(__builtin_amdgcn_s_wait_asynccnt)
  __builtin_amdgcn_s_wait_asynccnt(0);
#else
  asm volatile("s_wait_asynccnt 0x0" ::: "memory");
#endif
}

// ---------------- K1: B = A * A^T ----------------
__global__ __launch_bounds__(32) void k_aat(const float* __restrict__ A,
                                            float* __restrict__ B) {
  const int i0 = blockIdx.x * 16;          // M tile
  const int j0 = blockIdx.y * 16;          // N tile
  const int lane = threadIdx.x;
  const int l16 = lane & 15;
  const int koff = (lane < 16) ? 0 : 2;
  const float* ar = A + (size_t)(i0 + l16) * SZ;   // A-operand rows (M)
  const float* br = A + (size_t)(j0 + l16) * SZ;   // B-operand rows (N), K contiguous
  v8f acc = {0.f, 0.f, 0.f, 0.f, 0.f, 0.f, 0.f, 0.f};
#pragma unroll 4
  for (int k = 0; k < SZ; k += 4) {
    v2f a = *(const v2f*)(ar + k + koff);
    v2f b = *(const v2f*)(br + k + koff);
    acc = wmma4(a, b, acc);
  }
  const int mbase = i0 + ((lane < 16) ? 0 : 8);
  const int n = j0 + l16;
#pragma unroll
  for (int v = 0; v < 8; ++v) B[(size_t)(mbase + v) * SZ + n] = acc[v];
}

// ---------------- K2: inv2diag[i] = 2 / B[i,i] ----------------
__global__ __launch_bounds__(256) void k_diag(const float* __restrict__ B,
                                              float* __restrict__ inv2) {
  const int i = blockIdx.x * 256 + threadIdx.x;
  inv2[i] = 2.0f / B[(size_t)i * (SZ + 1)];
}

// ---------------- K3: p = A (float4 copy) ----------------
__global__ __launch_bounds__(256) void k_copy4(const float4* __restrict__ s,
                                               float4* __restrict__ d) {
  const int i = blockIdx.x * 256 + threadIdx.x;
  d[i] = s[i];
}

// ---------------- K4: sequential 16-row diagonal block solve ----------------
// Thread t owns column t; rows interact only within a column -> no barriers.
__global__ __launch_bounds__(1024) void k_block_solve(float* __restrict__ p,
                                                      const float* __restrict__ B,
                                                      const float* __restrict__ inv2,
                                                      int kb) {
  const int t = threadIdx.x;
  const int r0 = kb * 16;
  float rv[16];
#pragma unroll
  for (int r = 0; r < 16; ++r) rv[r] = p[(size_t)(r0 + r) * SZ + t];
#pragma unroll
  for (int i = 0; i < 15; ++i) {
#pragma unroll
    for (int j = i + 1; j < 16; ++j) {
      const float coef = B[(size_t)(r0 + i) * SZ + (r0 + j)] * inv2[r0 + j];
      rv[j] -= coef * rv[i];
    }
  }
#pragma unroll
  for (int r = 0; r < 16; ++r) p[(size_t)(r0 + r) * SZ + t] = rv[r];
}

// ---------------- K5: trailing update p[j,:] -= C[j,blk] * p[blk,:] ----------------
// A-operand = -inv2diag[j] * B[i,j] (sign folded in: f32 WMMA has only C-negate).
// C accumulator preloaded with p_old tile -> D = p_new.
__global__ __launch_bounds__(32) void k_trail(float* __restrict__ p,
                                              const float* __restrict__ B,
                                              const float* __restrict__ inv2,
                                              int kb) {
  const int i0 = kb * 16;                         // source block rows (K)
  const int j0 = (kb + 1) * 16 + blockIdx.x * 16; // dest row tile (M)
  const int n0 = blockIdx.y * 16;                 // col tile (N)
  const int lane = threadIdx.x;
  const int l16 = lane & 15;
  const int koff = (lane < 16) ? 0 : 2;
  const int mbase = j0 + ((lane < 16) ? 0 : 8);

  v8f acc;
#pragma unroll
  for (int v = 0; v < 8; ++v) acc[v] = p[(size_t)(mbase + v) * SZ + n0 + l16];

  const int m = j0 + l16;
  const float sa = -inv2[m];
#pragma unroll
  for (int kk = 0; kk < 16; kk += 4) {
    const int kr = i0 + kk + koff;
    v2f a, b;
    a.x = sa * B[(size_t)kr * SZ + m];
    a.y = sa * B[(size_t)(kr + 1) * SZ + m];
    b.x = p[(size_t)kr * SZ + n0 + l16];
    b.y = p[(size_t)(kr + 1) * SZ + n0 + l16];
    acc = wmma4(a, b, acc);
  }
#pragma unroll
  for (int v = 0; v < 8; ++v) p[(size_t)(mbase + v) * SZ + n0 + l16] = acc[v];
}

// ---------------- K6: pT = p^T (so phase-2 B-operand loads are b64-contiguous) ----
__global__ __launch_bounds__(256) void k_transpose(const float* __restrict__ p,
                                                   float* __restrict__ pT) {
  __shared__ float t[16][17];
  const int k0 = blockIdx.x * 16;
  const int n0 = blockIdx.y * 16;
  const int tx = threadIdx.x & 15;
  const int ty = threadIdx.x >> 4;
  t[ty][tx] = p[(size_t)(k0 + ty) * SZ + (n0 + tx)];
  __syncthreads();
  pT[(size_t)(n0 + ty) * SZ + (k0 + tx)] = t[tx][ty];
}

// ---------------- K7: fused main pass ----------------
// Per WG: 16 rows of x. Phase1: Bxs = (x*A^T) * inv2diag into LDS.
// Phase2: out = x - Bxs*p + b  (p via pT). 16 waves x 64 columns each.
// x tile staged via CDNA5 async global->LDS (ASYNCcnt), padded row stride.
__global__ __launch_bounds__(512) void k_main(const float* __restrict__ x,
                                              const float* __restrict__ A,
                                              const float* __restrict__ pT,
                                              const float* __restrict__ inv2,
                                              const float* __restrict__ bias,
                                              float* __restrict__ out) {
  __shared__ float sx[16 * SXP];
  __shared__ float sb[16 * SXP];
  const int row0 = blockIdx.x * 16;
  const int tid = threadIdx.x;

  // stage 16x1024 x-tile into LDS (padded rows) with async b128 copies
  {
    const float* src = x + (size_t)row0 * SZ;
#pragma unroll
    for (int it = 0; it < 8; ++it) {
      const int idx = tid + it * 512;     // 0..4095 float4
      const int r = idx >> 8;             // 256 float4 per row
      const int c4 = idx & 255;
      async_copy_b128(sx + r * SXP + c4 * 4, src + idx * 4);
    }
  }
  wait_async0();
  __syncthreads();

  const int wave = tid >> 5;
  const int lane = tid & 31;
  const int l16 = lane & 15;
  const int koff = (lane < 16) ? 0 : 2;
  const int hi8 = (lane < 16) ? 0 : 8;
  const int n0 = wave * 64;

  // ---- Phase 1: Bx tile for columns [n0, n0+64) ----
  v8f acc0 = {0,0,0,0,0,0,0,0}, acc1 = {0,0,0,0,0,0,0,0};
  v8f acc2 = {0,0,0,0,0,0,0,0}, acc3 = {0,0,0,0,0,0,0,0};
  const float* Ar0 = A + (size_t)(n0 +  0 + l16) * SZ;
  const float* Ar1 = A + (size_t)(n0 + 16 + l16) * SZ;
  const float* Ar2 = A + (size_t)(n0 + 32 + l16) * SZ;
  const float* Ar3 = A + (size_t)(n0 + 48 + l16) * SZ;
  const float* xr = sx + l16 * SXP;
#pragma unroll 2
  for (int k = 0; k < SZ; k += 4) {
    const v2f av = *(const v2f*)(xr + k + koff);   // shared A-operand (LDS)
    acc0 = wmma4(av, *(const v2f*)(Ar0 + k + koff), acc0);
    acc1 = wmma4(av, *(const v2f*)(Ar1 + k + koff), acc1);
    acc2 = wmma4(av, *(const v2f*)(Ar2 + k + koff), acc2);
    acc3 = wmma4(av, *(const v2f*)(Ar3 + k + koff), acc3);
  }
  {
    const float s0 = inv2[n0 +  0 + l16];
    const float s1 = inv2[n0 + 16 + l16];
    const float s2 = inv2[n0 + 32 + l16];
    const float s3 = inv2[n0 + 48 + l16];
#pragma unroll
    for (int v = 0; v < 8; ++v) {
      const int m = hi8 + v;
      sb[m * SXP + n0 +  0 + l16] = acc0[v] * s0;
      sb[m * SXP + n0 + 16 + l16] = acc1[v] * s1;
      sb[m * SXP + n0 + 32 + l16] = acc2[v] * s2;
      sb[m * SXP + n0 + 48 + l16] = acc3[v] * s3;
    }
  }
  __syncthreads();

  // ---- Phase 2: G = Bxs * p ; out = x - G + b ----
  v8f g0 = {0,0,0,0,0,0,0,0}, g1 = {0,0,0,0,0,0,0,0};
  v8f g2 = {0,0,0,0,0,0,0,0}, g3 = {0,0,0,0,0,0,0,0};
  const float* P0 = pT + (size_t)(n0 +  0 + l16) * SZ;
  const float* P1 = pT + (size_t)(n0 + 16 + l16) * SZ;
  const float* P2 = pT + (size_t)(n0 + 32 + l16) * SZ;
  const float* P3 = pT + (size_t)(n0 + 48 + l16) * SZ;
  const float* br = sb + l16 * SXP;
#pragma unroll 2
  for (int k = 0; k < SZ; k += 4) {
    const v2f av = *(const v2f*)(br + k + koff);
    g0 = wmma4(av, *(const v2f*)(P0 + k + koff), g0);
    g1 = wmma4(av, *(const v2f*)(P1 + k + koff), g1);
    g2 = wmma4(av, *(const v2f*)(P2 + k + koff), g2);
    g3 = wmma4(av, *(const v2f*)(P3 + k + koff), g3);
  }
  {
    const float b0 = bias[n0 +  0 + l16];
    const float b1 = bias[n0 + 16 + l16];
    const float b2 = bias[n0 + 32 + l16];
    const float b3 = bias[n0 + 48 + l16];
#pragma unroll
    for (int v = 0; v < 8; ++v) {
      const int m = hi8 + v;
      float* orow = out + (size_t)(row0 + m) * SZ;
      orow[n0 +  0 + l16] = sx[m * SXP + n0 +  0 + l16] - g0[v] + b0;
      orow[n0 + 16 + l16] = sx[m * SXP + n0 + 16 + l16] - g1[v] + b1;
      orow[n0 + 32 + l16] = sx[m * SXP + n0 + 32 + l16] - g2[v] + b2;
      orow[n0 + 48 + l16] = sx[m * SXP + n0 + 48 + l16] - g3[v] + b3;
    }
  }
}

extern "C" void kernel_launch(void* const* d_in, const int* in_sizes, int n_in,
                              void* d_out, int out_size, void* d_ws, size_t ws_size,
                              hipStream_t stream) {
  const float* x = (const float*)d_in[0];
  const float* A = (const float*)d_in[1];
  const float* bias = (const float*)d_in[2];
  float* out = (float*)d_out;

  float* ws = (float*)d_ws;
  float* B    = ws;                        // SZ*SZ floats
  float* p    = ws + (size_t)SZ * SZ;      // SZ*SZ floats
  float* inv2 = ws + (size_t)2 * SZ * SZ;  // SZ floats
  float* pT   = B;                         // B dead after elimination; reuse

  // Precalc (A-only): B = A*A^T, inv2diag, blocked forward substitution for p.
  k_aat<<<dim3(SZ / 16, SZ / 16), 32, 0, stream>>>(A, B);
  k_diag<<<SZ / 256, 256, 0, stream>>>(B, inv2);
  k_copy4<<<(SZ * SZ) / (256 * 4), 256, 0, stream>>>((const float4*)A, (float4*)p);
  for (int kb = 0; kb < SZ / 16; ++kb) {
    k_block_solve<<<1, 1024, 0, stream>>>(p, B, inv2, kb);
    const int mt = SZ / 16 - 1 - kb;
    if (mt > 0) k_trail<<<dim3(mt, SZ / 16), 32, 0, stream>>>(p, B, inv2, kb);
  }
  k_transpose<<<dim3(SZ / 16, SZ / 16), 256, 0, stream>>>(p, pT);

  // Fused main pass over all 65536 rows.
  const int nrows = in_sizes[0] / SZ;
  k_main<<<nrows / 16, 512, 0, stream>>>(x, A, pT, inv2, bias, out);
}